// Neighbor2PointEmbedding_63702954934710
// MI455X (gfx1250) — compile-verified
//
#include <hip/hip_runtime.h>
#include <math.h>

// Problem constants (match reference)
#define B_    16
#define N_    2048
#define KNB   32
#define COUT  64
#define HEADS 8
#define DEPTH 8

typedef __attribute__((ext_vector_type(2))) float v2f;
typedef __attribute__((ext_vector_type(8))) float v8f;

#define CHUNK  512               // columns scored per LDS pass
#define NCHUNK (N_ / CHUNK)      // 4
#define POOL   (NCHUNK * KNB)    // 128 candidates per row
#define WAVES  8                 // waves per KNN block
#define BIGF   3.0e38f

// ---------------------------------------------------------------------------
// Kernel 1: KNN via V_WMMA_F32_16X16X4_F32 Gram tiles + wave-parallel top-32.
// 8 waves (256 threads) per block; block owns 16 query rows of one batch.
//   Phase A: 32 column tiles/chunk split 4-per-wave (each wave does WMMAs).
//   Phase B: per-chunk top-32 extraction, 2 rows per wave (no intra-loop
//            block barriers; per-wave LDS ordering + volatile accesses).
//   Phase C: merge 4x32 chunk candidates -> 32 ascending, 2 rows per wave.
// ---------------------------------------------------------------------------
__global__ __launch_bounds__(256)
void knn_kernel(const float* __restrict__ coord, int* __restrict__ idx_out) {
  __shared__ float S[16 * CHUNK];   // 32 KB score tile (16 rows x 512 cols)
  __shared__ float Td[16 * POOL];   //  8 KB candidate distances
  __shared__ int   Ti[16 * POOL];   //  8 KB candidate indices

  const int b    = blockIdx.y;
  const int i0   = blockIdx.x * 16;
  const int wave = threadIdx.x >> 5;
  const int lane = threadIdx.x & 31;
  const int half = lane >> 4;       // 0 = K{0,1}, 1 = K{2,3-pad}
  const int l15  = lane & 15;
  const float* cb = coord + (size_t)b * 3 * N_;

  volatile float* Sv  = S;          // selection-phase accesses must not cache
  volatile float* Tdv = Td;

  // A tile: 16 queries x K4 (K=3 padded with zero), ISA 32-bit A layout:
  // lanes 0-15: row M=lane, VGPR0=K0, VGPR1=K1 ; lanes 16-31: K2, K3(=0)
  v2f A;
  A.x = cb[(half ? 2 : 0) * N_ + i0 + l15];
  A.y = half ? 0.0f : cb[1 * N_ + i0 + l15];

  for (int ch = 0; ch < NCHUNK; ++ch) {
    const int base = ch * CHUNK;

    // ---- Phase A: score 512 candidate columns; 4 WMMA tiles per wave ----
    for (int tt = 0; tt < (CHUNK / 16) / WAVES; ++tt) {
      const int t  = wave * ((CHUNK / 16) / WAVES) + tt;
      const int j0 = base + t * 16;
      v2f Bv;                      // B tile 4x16: col N=l15, same K split as A
      Bv.x = cb[(half ? 2 : 0) * N_ + j0 + l15];
      Bv.y = half ? 0.0f : cb[1 * N_ + j0 + l15];

      // |c_j|^2 for this lane's column: combine the two K-halves
      float part = Bv.x * Bv.x + Bv.y * Bv.y;
      float sqj  = part + __shfl_xor(part, 16, 32);

      v8f acc = {0.f, 0.f, 0.f, 0.f, 0.f, 0.f, 0.f, 0.f};
      acc = __builtin_amdgcn_wmma_f32_16x16x4_f32(
          /*neg_a=*/false, A, /*neg_b=*/false, Bv,
          /*c_mod=*/(short)0, acc, /*reuse_a=*/false, /*reuse_b=*/false);

      // D layout: VGPR r -> rows r (lanes 0-15) / r+8 (lanes 16-31), col=l15.
      // score = |c_j|^2 - 2 c_i.c_j (row-constant |c_i|^2 dropped: same order)
#pragma unroll
      for (int r = 0; r < 8; ++r) {
        const int row = half ? (r + 8) : r;
        S[row * CHUNK + t * 16 + l15] = sqj - 2.0f * acc[r];
      }
    }
    __syncthreads();

    // ---- Phase B: per-row top-32 extraction; wave owns rows 2w, 2w+1 ----
    for (int mm = 0; mm < 2; ++mm) {
      const int m = wave * 2 + mm;
      for (int rr = 0; rr < KNB; ++rr) {
        float bv = BIGF; int bj = 0x7fffffff;
#pragma unroll
        for (int t = 0; t < CHUNK / 32; ++t) {
          const int col = lane + t * 32;
          const float v = Sv[m * CHUNK + col];
          const int   j = base + col;
          if (v < bv || (v == bv && j < bj)) { bv = v; bj = j; }
        }
#pragma unroll
        for (int off = 16; off >= 1; off >>= 1) {   // lexicographic argmin
          const float ov = __shfl_xor(bv, off, 32);
          const int   oj = __shfl_xor(bj, off, 32);
          if (ov < bv || (ov == bv && oj < bj)) { bv = ov; bj = oj; }
        }
        if (lane == 0) {
          Sv[m * CHUNK + (bj - base)]  = BIGF;      // mark consumed
          Td[m * POOL + ch * KNB + rr] = bv;
          Ti[m * POOL + ch * KNB + rr] = bj;
        }
        // same-wave DS ops complete in order; no block barrier needed here
      }
    }
    __syncthreads();   // protect S reuse by next chunk's Phase A
  }

  // ---- Phase C: merge 128 candidates per row -> 32 ascending indices ----
  for (int mm = 0; mm < 2; ++mm) {
    const int m = wave * 2 + mm;                    // same wave wrote Td/Ti[m]
    int* orow = idx_out + ((size_t)b * N_ + (i0 + m)) * KNB;
    for (int rr = 0; rr < KNB; ++rr) {
      float bv = BIGF; int bj = 0x7fffffff; int bs = 0;
#pragma unroll
      for (int t = 0; t < POOL / 32; ++t) {
        const int slot = lane + t * 32;
        const float v  = Tdv[m * POOL + slot];
        const int   j  = Ti[m * POOL + slot];
        if (v < bv || (v == bv && j < bj)) { bv = v; bj = j; bs = slot; }
      }
#pragma unroll
      for (int off = 16; off >= 1; off >>= 1) {
        const float ov = __shfl_xor(bv, off, 32);
        const int   oj = __shfl_xor(bj, off, 32);
        const int   os = __shfl_xor(bs, off, 32);
        if (ov < bv || (ov == bv && oj < bj)) { bv = ov; bj = oj; bs = os; }
      }
      if (lane == 0) {
        Tdv[m * POOL + bs] = BIGF;
        orow[rr] = bj;
      }
    }
  }
}

// ---------------------------------------------------------------------------
// Kernel 2: gather-diff + q/k/v projection + 8-head attention.
// One wave per point; lane == neighbor (K_NBRS == 32 == wave32).
// ---------------------------------------------------------------------------
__global__ __launch_bounds__(128)
void attn_kernel(const float* __restrict__ x,
                 const float* __restrict__ qw,
                 const float* __restrict__ kw,
                 const float* __restrict__ vw,
                 const int*   __restrict__ knn,
                 float*       __restrict__ out) {
  const int wave = threadIdx.x >> 5;
  const int lane = threadIdx.x & 31;
  const int p = blockIdx.x * 4 + wave;    // global point id, 0..B*N-1
  const int b = p / N_;
  const int n = p % N_;

  const float* xb = x + (size_t)b * 3 * N_;
  const int j = knn[(size_t)p * KNB + lane];

  const float xc0 = xb[0 * N_ + n], xc1 = xb[1 * N_ + n], xc2 = xb[2 * N_ + n];
  const float d0 = xb[0 * N_ + j] - xc0;   // 'diff' grouping
  const float d1 = xb[1 * N_ + j] - xc1;
  const float d2 = xb[2 * N_ + j] - xc2;

  const float scale = 0.35355339059327373f;   // 1/sqrt(8)
  float* ob = out + (size_t)b * COUT * N_ + n;

  for (int h = 0; h < HEADS; ++h) {
    float vv[DEPTH];
    float e = 0.f;
#pragma unroll
    for (int d = 0; d < DEPTH; ++d) {
      const int o = h * DEPTH + d;
      const float q = qw[o*3+0]*xc0 + qw[o*3+1]*xc1 + qw[o*3+2]*xc2;
      const float k = kw[o*3+0]*d0  + kw[o*3+1]*d1  + kw[o*3+2]*d2;
      vv[d]         = vw[o*3+0]*d0  + vw[o*3+1]*d1  + vw[o*3+2]*d2;
      e += q * k;
    }
    e *= scale;

    // softmax across the 32 neighbor lanes
    float mx = e;
#pragma unroll
    for (int off = 16; off >= 1; off >>= 1) mx = fmaxf(mx, __shfl_xor(mx, off, 32));
    const float pr = __expf(e - mx);
    float sm = pr;
#pragma unroll
    for (int off = 16; off >= 1; off >>= 1) sm += __shfl_xor(sm, off, 32);
    const float a = pr / sm;

    float acc[DEPTH];
#pragma unroll
    for (int d = 0; d < DEPTH; ++d) {
      float s = a * vv[d];
#pragma unroll
      for (int off = 16; off >= 1; off >>= 1) s += __shfl_xor(s, off, 32);
      acc[d] = s;
    }
    if (lane == 0) {
#pragma unroll
      for (int d = 0; d < DEPTH; ++d) ob[(h * DEPTH + d) * N_] = acc[d];
    }
  }
}

// ---------------------------------------------------------------------------
extern "C" void kernel_launch(void* const* d_in, const int* in_sizes, int n_in,
                              void* d_out, int out_size, void* d_ws, size_t ws_size,
                              hipStream_t stream) {
  (void)in_sizes; (void)n_in; (void)out_size; (void)ws_size;
  const float* x  = (const float*)d_in[0];
  const float* co = (const float*)d_in[1];
  const float* qw = (const float*)d_in[2];
  const float* kw = (const float*)d_in[3];
  const float* vw = (const float*)d_in[4];
  float* out = (float*)d_out;
  int*   knn = (int*)d_ws;   // B*N*KNB ints = 4 MB scratch

  knn_kernel <<<dim3(N_ / 16, B_), 256, 0, stream>>>(co, knn);
  attn_kernel<<<(B_ * N_) / 4, 128, 0, stream>>>(x, qw, kw, vw, knn, out);
}